// MovieCoreference_48034914238593
// MI455X (gfx1250) — compile-verified
//
#include <hip/hip_runtime.h>
#include <hip/hip_bf16.h>

typedef float v2f __attribute__((ext_vector_type(2)));
typedef float v8f __attribute__((ext_vector_type(8)));
typedef unsigned int v4u __attribute__((ext_vector_type(4)));
typedef int v4i __attribute__((ext_vector_type(4)));
typedef int v8i __attribute__((ext_vector_type(8)));

#define NEGV -1000000000.0f
#define E_DIM 768
#define N_WORDS 1024
#define K_TOP 50
#define M_ROWS (N_WORDS * K_TOP)   // 51200
#define NBLK (E_DIM / 128)          // 6 N-blocks for the pair GEMM

// -------------------------------------------------------------------------
// Tensor Data Mover: contiguous 48KB (16 x 768 f32) global -> LDS copy.
// D# built per CDNA5 ISA ch.8: group0 = {count|flags, lds_addr,
// global_addr[31:0], global_addr[56:32]|type=2}; group1 carries data_size=4B,
// tensor_dim0 = tile_dim0 = 12288 elements (1-D tile), stride 12288.
// Issue from one wave, complete with s_wait_tensorcnt, publish via barrier.
// -------------------------------------------------------------------------
__device__ __forceinline__ void tdm_load_tile48k(const float* __restrict__ gsrc,
                                                 float* lds_dst) {
    unsigned long long ga = (unsigned long long)(uintptr_t)gsrc;
    unsigned int ldsoff = (unsigned int)(uintptr_t)lds_dst;  // flat addr low32 = LDS offset
    v4u g0;
    g0.x = 1u;                                               // count=1 (valid), user mode
    g0.y = ldsoff;                                           // lds_addr (bytes)
    g0.z = (unsigned int)ga;                                 // global_addr[31:0]
    g0.w = ((unsigned int)(ga >> 32) & 0x01FFFFFFu) | (2u << 30); // addr[56:32] | type=2
    v8i g1;
    g1[0] = 0x00020000;        // workgroup_mask=0, data_size=2 (4 bytes)
    g1[1] = 0x30000000;        // tensor_dim0[15:0]=12288 at bits[63:48]
    g1[2] = 0x00010000;        // tensor_dim0 hi=0, tensor_dim1=1
    g1[3] = 0x30000000;        // tile_dim0=12288 at bits[127:112]
    g1[4] = 0;                 // tile_dim1=0, tile_dim2=0 (1-D tile)
    g1[5] = 12288;             // tensor_dim0_stride
    g1[6] = 0;
    g1[7] = 0;
    v4i z4 = {0, 0, 0, 0};
#if __has_include(<hip/amd_detail/amd_gfx1250_TDM.h>)
    v8i z8 = {0, 0, 0, 0, 0, 0, 0, 0};
    __builtin_amdgcn_tensor_load_to_lds(g0, g1, z4, z4, z8, 0);
#else
    __builtin_amdgcn_tensor_load_to_lds(g0, g1, z4, z4, 0);
#endif
}

// -------------------------------------------------------------------------
// fp32 WMMA inner loop: one wave accumulates a 16x16 tile over K=768.
// A tile (16 x 768) resides in LDS. BT=false: B row-major KxN (ldb=N).
// BT=true: B row-major NxK (ldb=768).
// A layout: lane m=lane&15 holds k-pair (lane>>4)*2; C/D: m=v+8*(lane>>4).
// -------------------------------------------------------------------------
template <bool BT>
__device__ __forceinline__ v8f wmma_row_tile(const float* __restrict__ As,
                                             const float* __restrict__ B,
                                             int ncol, int ldb) {
    const int lane  = threadIdx.x & 31;
    const int mrow  = lane & 15;
    const int khalf = (lane >> 4) << 1;   // 0 or 2
    v8f acc = {};
    #pragma unroll 4
    for (int k0 = 0; k0 < E_DIM; k0 += 4) {
        v2f a;
        a.x = As[mrow * E_DIM + k0 + khalf];
        a.y = As[mrow * E_DIM + k0 + khalf + 1];
        v2f b;
        if (BT) {
            b.x = B[ncol * ldb + k0 + khalf];
            b.y = B[ncol * ldb + k0 + khalf + 1];
        } else {
            b.x = B[(k0 + khalf) * ldb + ncol];
            b.y = B[(k0 + khalf + 1) * ldb + ncol];
        }
        acc = __builtin_amdgcn_wmma_f32_16x16x4_f32(
            false, a, false, b, (short)0, acc, false, false);
    }
    return acc;
}

// -------------------------------------------------------------------------
// C[M,N] = A[M,768] @ B[768,N] (+ bias). grid = (M/16, N/128), block 256.
// -------------------------------------------------------------------------
__global__ __launch_bounds__(256) void gemm_nn_kernel(
    const float* __restrict__ A, const float* __restrict__ B,
    const float* __restrict__ bias, float* __restrict__ C, int N) {
    __shared__ __align__(16) float As[16 * E_DIM];
    const int row_base = blockIdx.x * 16;
    if (threadIdx.x < 32) {   // one wave issues the TDM transfer
        tdm_load_tile48k(A + (size_t)row_base * E_DIM, As);
        __builtin_amdgcn_s_wait_tensorcnt(0);
    }
    __syncthreads();

    const int wave = threadIdx.x >> 5;
    const int lane = threadIdx.x & 31;
    const int ncol = blockIdx.y * 128 + wave * 16 + (lane & 15);
    v8f acc = wmma_row_tile<false>(As, B, ncol, N);

    const float bv = bias ? bias[ncol] : 0.0f;
    const int hi = lane >> 4;
    #pragma unroll
    for (int v = 0; v < 8; ++v) {
        int m = v + 8 * hi;
        C[(size_t)(row_base + m) * N + ncol] = acc[v] + bv;
    }
}

// coarse = proj @ WE^T + cs[i] + cs[j], masked to NEG where !(j < i)
__global__ __launch_bounds__(256) void gemm_coarse_kernel(
    const float* __restrict__ proj, const float* __restrict__ WE,
    const float* __restrict__ cs, float* __restrict__ C) {
    __shared__ __align__(16) float As[16 * E_DIM];
    const int row_base = blockIdx.x * 16;
    if (threadIdx.x < 32) {
        tdm_load_tile48k(proj + (size_t)row_base * E_DIM, As);
        __builtin_amdgcn_s_wait_tensorcnt(0);
    }
    __syncthreads();

    const int wave = threadIdx.x >> 5;
    const int lane = threadIdx.x & 31;
    const int ncol = blockIdx.y * 128 + wave * 16 + (lane & 15);
    v8f acc = wmma_row_tile<true>(As, WE, ncol, E_DIM);

    const float csj = cs[ncol];
    const int hi = lane >> 4;
    #pragma unroll
    for (int v = 0; v < 8; ++v) {
        int m = v + 8 * hi;
        int row = row_base + m;
        float val = acc[v] + cs[row] + csj;
        C[(size_t)row * N_WORDS + ncol] = (ncol < row) ? val : NEGV;
    }
}

// -------------------------------------------------------------------------
// Feature tables: gconst[e] = b1[e] + genre_emb[gid]@W1[3E:3E+20]
//                 dtab[9][e] = dist_emb @ W1[3E+20:3E+40]
//                 stab[2][e] = speaker_emb @ W1[3E+40:3E+60]
// -------------------------------------------------------------------------
__global__ __launch_bounds__(256) void tables_kernel(
    const float* __restrict__ W1, const float* __restrict__ b1,
    const float* __restrict__ genre_emb, const float* __restrict__ dist_emb,
    const float* __restrict__ spk_emb, const int* __restrict__ genre_id,
    float* __restrict__ gconst, float* __restrict__ dtab, float* __restrict__ stab) {
    int e = blockIdx.x * blockDim.x + threadIdx.x;
    if (e >= E_DIM) return;
    const int base = 3 * E_DIM;
    int gid = genre_id[0];
    float g = b1[e];
    for (int t = 0; t < 20; ++t) g += genre_emb[gid * 20 + t] * W1[(size_t)(base + t) * E_DIM + e];
    gconst[e] = g;
    for (int d = 0; d < 9; ++d) {
        float s = 0.f;
        for (int t = 0; t < 20; ++t) s += dist_emb[d * 20 + t] * W1[(size_t)(base + 20 + t) * E_DIM + e];
        dtab[d * E_DIM + e] = s;
    }
    for (int sp = 0; sp < 2; ++sp) {
        float s = 0.f;
        for (int t = 0; t < 20; ++t) s += spk_emb[sp * 20 + t] * W1[(size_t)(base + 40 + t) * E_DIM + e];
        stab[sp * E_DIM + e] = s;
    }
}

// -------------------------------------------------------------------------
// Top-50 per row (descending, lowest-index tie-break, matching lax.top_k)
// -------------------------------------------------------------------------
__global__ __launch_bounds__(256) void topk_kernel(
    const float* __restrict__ coarse, float* __restrict__ top_scores,
    int* __restrict__ top_idx) {
    __shared__ float sv[N_WORDS];
    __shared__ float rv[256];
    __shared__ int ri[256];
    const int i = blockIdx.x;
    for (int j = threadIdx.x; j < N_WORDS; j += 256)
        sv[j] = coarse[(size_t)i * N_WORDS + j];
    __syncthreads();
    for (int it = 0; it < K_TOP; ++it) {
        float best = -3.4e38f;
        int bidx = 1 << 30;
        for (int j = threadIdx.x; j < N_WORDS; j += 256) {
            float v = sv[j];
            if (v > best || (v == best && j < bidx)) { best = v; bidx = j; }
        }
        rv[threadIdx.x] = best; ri[threadIdx.x] = bidx;
        __syncthreads();
        for (int s = 128; s > 0; s >>= 1) {
            if (threadIdx.x < s) {
                float ov = rv[threadIdx.x + s]; int oi = ri[threadIdx.x + s];
                if (ov > rv[threadIdx.x] || (ov == rv[threadIdx.x] && oi < ri[threadIdx.x])) {
                    rv[threadIdx.x] = ov; ri[threadIdx.x] = oi;
                }
            }
            __syncthreads();
        }
        if (threadIdx.x == 0) {
            top_scores[i * K_TOP + it] = rv[0];
            top_idx[i * K_TOP + it] = ri[0];
            sv[ri[0]] = -3.4e38f;
        }
        __syncthreads();
    }
}

// -------------------------------------------------------------------------
// Fused pair kernel. Row m = i*50 + k. Builds ab = WE[i]*WE[t] tile in LDS,
// WMMA against W1[2E:3E], epilogue adds projB[i]+projA[t]+gconst+dtab+stab,
// leaky-ReLU, then deterministic (atomic-free) reduction of h*W2 into
// ana_part[blockIdx.y][m]. grid: (3200, 6), block 256.
// -------------------------------------------------------------------------
__global__ __launch_bounds__(256) void pair_kernel(
    const float* __restrict__ WE, const float* __restrict__ W1ab,
    const float* __restrict__ projB, const float* __restrict__ projA,
    const float* __restrict__ gconst, const float* __restrict__ dtab,
    const float* __restrict__ stab, const float* __restrict__ W2,
    const int* __restrict__ top_idx, const int* __restrict__ spk_ids,
    float* __restrict__ ana_part) {
    __shared__ __align__(16) float As[16 * E_DIM];
    __shared__ float s_hw[8 * 256];     // per-wave 16x16 h*w2 products
    __shared__ int s_i[16], s_t[16], s_db[16], s_sp[16];

    const int mbase = blockIdx.x * 16;
    const int tid = threadIdx.x;
    if (tid < 16) {
        int m = mbase + tid;
        int i = m / K_TOP;
        int t = top_idx[m];
        s_i[tid] = i; s_t[tid] = t;
        int d = i - t; if (d < 1) d = 1;
        int lg = 31 - __clz(d); if (lg > 6) lg = 6;
        s_db[tid] = (d < 5) ? (d - 1) : (lg + 2);
        s_sp[tid] = (spk_ids[i] == spk_ids[t]) ? 1 : 0;
    }
    __syncthreads();

    {   // build a*b tile (computed, so staged by VALU rather than TDM)
        const float4* WE4 = (const float4*)WE;
        float4* As4 = (float4*)As;
        for (int idx = tid; idx < 16 * (E_DIM / 4); idx += 256) {
            int r = idx / (E_DIM / 4);
            int c = idx % (E_DIM / 4);
            float4 x = WE4[(size_t)s_i[r] * (E_DIM / 4) + c];
            float4 y = WE4[(size_t)s_t[r] * (E_DIM / 4) + c];
            As4[idx] = make_float4(x.x * y.x, x.y * y.y, x.z * y.z, x.w * y.w);
        }
    }
    __syncthreads();

    const int wave = tid >> 5;
    const int lane = tid & 31;
    const int nloc = lane & 15;
    const int ncol = blockIdx.y * 128 + wave * 16 + nloc;
    v8f acc = wmma_row_tile<false>(As, W1ab, ncol, E_DIM);

    const float w2 = W2[ncol];
    const float gc = gconst[ncol];
    const int hi = lane >> 4;
    #pragma unroll
    for (int v = 0; v < 8; ++v) {
        int m = v + 8 * hi;
        float add = projB[(size_t)s_i[m] * E_DIM + ncol]
                  + projA[(size_t)s_t[m] * E_DIM + ncol]
                  + gc
                  + dtab[s_db[m] * E_DIM + ncol]
                  + stab[s_sp[m] * E_DIM + ncol];
        float h = acc[v] + add;
        h = (h > 0.0f) ? h : 0.01f * h;          // leaky_relu slope 0.01
        s_hw[wave * 256 + m * 16 + nloc] = h * w2;
    }
    __syncthreads();

    if (tid < 16) {   // fixed-order, deterministic 128-way reduction per row
        float s = 0.0f;
        for (int w = 0; w < 8; ++w)
            #pragma unroll
            for (int n = 0; n < 16; ++n)
                s += s_hw[w * 256 + tid * 16 + n];
        ana_part[(size_t)blockIdx.y * M_ROWS + mbase + tid] = s;
    }
}

__global__ __launch_bounds__(256) void fine_kernel(
    const float* __restrict__ top_scores, const float* __restrict__ ana_part,
    const float* __restrict__ b2, float* __restrict__ out) {
    int m = blockIdx.x * 256 + threadIdx.x;
    if (m < M_ROWS) {
        float s = 0.0f;
        #pragma unroll
        for (int y = 0; y < NBLK; ++y) s += ana_part[(size_t)y * M_ROWS + m];
        float ts = top_scores[m];
        out[m] = (ts > NEGV * 0.5f) ? (ts + s + b2[0]) : NEGV;
    }
}

// -------------------------------------------------------------------------
extern "C" void kernel_launch(void* const* d_in, const int* in_sizes, int n_in,
                              void* d_out, int out_size, void* d_ws, size_t ws_size,
                              hipStream_t stream) {
    const float* WE       = (const float*)d_in[0];
    const float* cs       = (const float*)d_in[1];
    const float* coarse_W = (const float*)d_in[2];
    const float* coarse_b = (const float*)d_in[3];
    const float* W1       = (const float*)d_in[4];
    const float* b1       = (const float*)d_in[5];
    const float* W2       = (const float*)d_in[6];
    const float* b2       = (const float*)d_in[7];
    const float* genre_e  = (const float*)d_in[8];
    const float* dist_e   = (const float*)d_in[9];
    const float* spk_e    = (const float*)d_in[10];
    const int*   spk_ids  = (const int*)d_in[11];
    const int*   genre_id = (const int*)d_in[12];
    float* out = (float*)d_out;

    // workspace carve-out (~15.5 MB)
    char* ws = (char*)d_ws;
    float* proj     = (float*)ws;  ws += (size_t)N_WORDS * E_DIM * 4;
    float* projB    = (float*)ws;  ws += (size_t)N_WORDS * E_DIM * 4;
    float* projA    = (float*)ws;  ws += (size_t)N_WORDS * E_DIM * 4;
    float* coarse   = (float*)ws;  ws += (size_t)N_WORDS * N_WORDS * 4;
    float* tscore   = (float*)ws;  ws += (size_t)M_ROWS * 4;
    float* ana_part = (float*)ws;  ws += (size_t)NBLK * M_ROWS * 4;
    float* gconst   = (float*)ws;  ws += (size_t)E_DIM * 4;
    float* dtab     = (float*)ws;  ws += (size_t)9 * E_DIM * 4;
    float* stab     = (float*)ws;  ws += (size_t)2 * E_DIM * 4;
    int*   tidx     = (int*)ws;    ws += (size_t)M_ROWS * 4;

    // 1) tiny feature tables
    tables_kernel<<<dim3(3), dim3(256), 0, stream>>>(
        W1, b1, genre_e, dist_e, spk_e, genre_id, gconst, dtab, stab);

    // 2) proj = WE @ coarse_W + coarse_b
    gemm_nn_kernel<<<dim3(N_WORDS / 16, E_DIM / 128), dim3(256), 0, stream>>>(
        WE, coarse_W, coarse_b, proj, E_DIM);

    // 3) projB = WE @ W1[0:E],  projA = WE @ W1[E:2E]
    gemm_nn_kernel<<<dim3(N_WORDS / 16, E_DIM / 128), dim3(256), 0, stream>>>(
        WE, W1, nullptr, projB, E_DIM);
    gemm_nn_kernel<<<dim3(N_WORDS / 16, E_DIM / 128), dim3(256), 0, stream>>>(
        WE, W1 + (size_t)E_DIM * E_DIM, nullptr, projA, E_DIM);

    // 4) coarse = proj @ WE^T + cs_i + cs_j, causal mask
    gemm_coarse_kernel<<<dim3(N_WORDS / 16, N_WORDS / 128), dim3(256), 0, stream>>>(
        proj, WE, cs, coarse);

    // 5) top-50 per row
    topk_kernel<<<dim3(N_WORDS), dim3(256), 0, stream>>>(coarse, tscore, tidx);

    // 6) fused pair GEMM + deterministic partial reduction
    pair_kernel<<<dim3(M_ROWS / 16, NBLK), dim3(256), 0, stream>>>(
        WE, W1 + (size_t)2 * E_DIM * E_DIM, projB, projA,
        gconst, dtab, stab, W2, tidx, spk_ids, ana_part);

    // 7) sum partials + final mask + bias
    fine_kernel<<<dim3((M_ROWS + 255) / 256), dim3(256), 0, stream>>>(
        tscore, ana_part, b2, out);
}